// _ROIAligner_3238405341320
// MI455X (gfx1250) — compile-verified
//
#include <hip/hip_runtime.h>

typedef __attribute__((ext_vector_type(2))) float v2f;
typedef __attribute__((ext_vector_type(8))) float v8f;
typedef int v4i __attribute__((vector_size(16)));          // matches builtin param pointee
typedef __attribute__((address_space(1))) v4i* g_v4i_ptr;  // global (device) int4*
typedef __attribute__((address_space(3))) v4i* l_v4i_ptr;  // LDS (shared) int4*

#define C_TOTAL 256
#define CCH 64          // channels per chunk
#define NCHUNK (C_TOTAL / CCH)
#define G196 196        // 14x14 gathered grid positions (permuted: pixel*4 + tap)

#if defined(__gfx1250__) && __has_builtin(__builtin_amdgcn_global_load_async_to_lds_b128)
#define USE_ASYNC 1
#else
#define USE_ASYNC 0
#endif

__global__ __launch_bounds__(128) void roi_align_wmma_f32(
    const float* __restrict__ f2, const float* __restrict__ f3,
    const float* __restrict__ f4, const float* __restrict__ f5,
    const float* __restrict__ boxes, float* __restrict__ out)
{
    __shared__ float kyS[14];
    __shared__ float kxS[14];
    __shared__ int   yiS[14];
    __shared__ int   xiS[14];
    __shared__ const float* rowp[G196];
    __shared__ __align__(16) float F[2][G196][CCH];   // double buffer, 2 x 50176 B

    const int t   = threadIdx.x;
    const int bid = blockIdx.x;            // 0..1023 = b*512 + n
    const int b   = bid >> 9;

    // ---- box -> level, scaled coords (uniform across block; redundant per thread) ----
    const float* bx = boxes + (size_t)bid * 4;
    float by0 = bx[0], bx0 = bx[1], by2 = bx[2], bx2 = bx[3];
    float bh = by2 - by0, bw = bx2 - bx0;
    float asq  = sqrtf(bh * bw);
    float lvlf = floorf(log2f(asq * (1.0f / 224.0f))) + 4.0f;
    if (!(lvlf >= 2.0f)) lvlf = 2.0f;      // also catches NaN / -inf
    if (lvlf > 5.0f)     lvlf = 5.0f;
    int lvl  = (int)lvlf;
    int lidx = lvl - 2;
    const float* fptr;
    int HL;                                 // level is square: H == W
    if      (lidx == 0) { fptr = f2; HL = 256; }
    else if (lidx == 1) { fptr = f3; HL = 128; }
    else if (lidx == 2) { fptr = f4; HL = 64;  }
    else                { fptr = f5; HL = 32;  }
    float inv   = 1.0f / (float)(1 << lvl);
    float y0s   = by0 * inv, x0s = bx0 * inv;
    float bhs   = bh  * inv, bws = bw  * inv;
    float bound = (float)(HL - 1);

    if (t < 7) {
        float fi  = ((float)t + 0.5f) * (1.0f / 7.0f);
        float gy  = y0s + fi * bhs;
        float gx  = x0s + fi * bws;
        float fy0 = fminf(fmaxf(floorf(gy), 0.0f), bound);
        float fy1 = fminf(fy0 + 1.0f, bound);
        float fx0 = fminf(fmaxf(floorf(gx), 0.0f), bound);
        float fx1 = fminf(fx0 + 1.0f, bound);
        float ly = gy - fy0, lx = gx - fx0;
        kyS[2*t]   = 1.0f - ly;  kyS[2*t+1] = ly;
        kxS[2*t]   = 1.0f - lx;  kxS[2*t+1] = lx;
        yiS[2*t]   = (int)fy0;   yiS[2*t+1] = (int)fy1;
        xiS[2*t]   = (int)fx0;   xiS[2*t+1] = (int)fx1;
    }
    __syncthreads();

    // permuted row pointers: g = kb*4 + r, kb = output pixel (sy*7+sx), r = 2x2 tap
    for (int g = t; g < G196; g += 128) {
        int kb = g >> 2, r = g & 3;
        int sy = kb / 7, sx = kb - sy * 7;
        int yy = yiS[2*sy + (r >> 1)];
        int xx = xiS[2*sx + (r & 1)];
        rowp[g] = fptr + (((size_t)b * HL + yy) * HL + xx) * C_TOTAL;
    }
    __syncthreads();

    const int lane = t & 31;
    const int wv   = t >> 5;      // wave id = N tile (16 channels within chunk)
    const int hi   = lane >> 4;   // 0: lanes 0-15, 1: lanes 16-31
    const int lm   = lane & 15;

    float* outbase = out + (size_t)bid * 49 * C_TOTAL;

    // ---- gather issue: 196 rows x 64 channels (16B per lane-op) ----
    auto issue_gather = [&](int ch, int bufi) {
        const int ch0 = ch * CCH;
        for (int idx = t; idx < G196 * (CCH / 4); idx += 128) {
            int g = idx >> 4;
            int q = idx & 15;
            const float* src = rowp[g] + ch0 + q * 4;
            float*       dst = &F[bufi][g][q * 4];
#if USE_ASYNC
            __builtin_amdgcn_global_load_async_to_lds_b128(
                (g_v4i_ptr)src, (l_v4i_ptr)dst, 0, 0);
#else
            *(float4*)dst = *(const float4*)src;
#endif
        }
    };

    auto wait_gather = [&]() {
#if USE_ASYNC
#if __has_builtin(__builtin_amdgcn_s_wait_asynccnt)
        __builtin_amdgcn_s_wait_asynccnt(0);
#else
        asm volatile("s_wait_asynccnt 0" ::: "memory");
#endif
#endif
    };

    issue_gather(0, 0);

    for (int ch = 0; ch < NCHUNK; ++ch) {
        const int cur = ch & 1;
        const int ch0 = ch * CCH;

        wait_gather();        // my async loads for F[cur] complete
        __syncthreads();      // everyone's loads landed; prev compute done

        if (ch + 1 < NCHUNK)
            issue_gather(ch + 1, cur ^ 1);   // overlaps with compute below

        // ---- WMMA combine: k-step kb feeds only output row m==kb (block-aligned W) ----
        for (int mt = 0; mt < 4; ++mt) {            // M tiles cover pixels 0..63 (49 used)
            v8f acc = {};
            int kbend = (mt == 3) ? 49 : (mt * 16 + 16);
            for (int kb = mt * 16; kb < kbend; ++kb) {
                int sy = kb / 7, sx = kb - sy * 7;
                // A fragment (16x4 f32): lane lm row, K = hi*2 + {0,1}
                float kyv = kyS[2*sy + hi];
                bool  on  = (lm == (kb & 15));
                v2f a;
                a.x = on ? kyv * kxS[2*sx]     : 0.0f;
                a.y = on ? kyv * kxS[2*sx + 1] : 0.0f;
                // B fragment (4x16 f32): K = hi*2 + {0,1}, N = lm
                int row = (kb << 2) + (hi << 1);
                v2f bf;
                bf.x = F[cur][row][wv * 16 + lm];
                bf.y = F[cur][row + 1][wv * 16 + lm];
                acc = __builtin_amdgcn_wmma_f32_16x16x4_f32(
                        false, a, false, bf, (short)0, acc, false, false);
            }
            // D layout: VGPR v -> M = v + hi*8, N = lm
            #pragma unroll
            for (int v = 0; v < 8; ++v) {
                int pix = mt * 16 + v + hi * 8;
                if (pix < 49)
                    outbase[(size_t)pix * C_TOTAL + ch0 + wv * 16 + lm] = acc[v];
            }
        }
    }
}

extern "C" void kernel_launch(void* const* d_in, const int* in_sizes, int n_in,
                              void* d_out, int out_size, void* d_ws, size_t ws_size,
                              hipStream_t stream) {
    (void)in_sizes; (void)n_in; (void)out_size; (void)d_ws; (void)ws_size;
    const float* f2 = (const float*)d_in[0];
    const float* f3 = (const float*)d_in[1];
    const float* f4 = (const float*)d_in[2];
    const float* f5 = (const float*)d_in[3];
    const float* bx = (const float*)d_in[4];
    float* out = (float*)d_out;
    hipLaunchKernelGGL(roi_align_wmma_f32, dim3(1024), dim3(128), 0, stream,
                       f2, f3, f4, f5, bx, out);
}